// SIDARTHEOde_63204738728592
// MI455X (gfx1250) — compile-verified
//
#include <hip/hip_runtime.h>

// SIDARTHE 8-state ODE, 3/8-rule RK4, 20000 steps, trajectory (20001 x 8) f32.
// Serial recurrence -> single wave32, one active lane, state in VGPRs.
// Output streamed via CDNA5 Tensor Data Mover: LDS ring (2 x 512 rows x 32B)
// flushed with tensor_store_from_lds, double-buffered on TENSORcnt.

#define STEPS      20000
#define NSTATE     8
#define SLOT_ROWS  512          // 16 KB per slot
#define SLOT_ELEMS (SLOT_ROWS * NSTATE)

typedef __attribute__((ext_vector_type(4))) unsigned int u32x4;
typedef __attribute__((ext_vector_type(8))) int          i32x8;
typedef __attribute__((ext_vector_type(4))) int          i32x4;

struct Params {
    float a, b, g, d;          // alpha beta gamma delta
    float eps, th, ze, et;     // epsilon theta zeta eta
    float m, n, ta, la;        // mu nu tau lambda
    float rh, ka, x, si;       // rho kappa xi sigma
    // hoisted loop-invariant sums (same values the reference recomputes per eval)
    float cD, cA, cR, cT, cI;  // (et+rh), (th+m+ka), (n+x), (si+ta), (eps+ze+la)
};

__device__ __forceinline__ void rhs(const float* __restrict__ y, const Params& P,
                                    float* __restrict__ dy)
{
    const float S = y[0], D = y[1], A = y[2], R = y[3], T = y[4], I = y[5];
    const float force = S * (P.a * I + P.b * D + P.g * A + P.d * R);
    dy[0] = force;
    dy[1] = P.eps * I - P.cD * D;
    dy[2] = P.ze  * I - P.cA * A;
    dy[3] = P.et * D + P.th * A - P.cR * R;
    dy[4] = P.m  * A + P.n  * R - P.cT * T;
    dy[5] = force - P.cI * I;
    dy[6] = P.la * I + P.rh * D + P.ka * A + P.x * R + P.si * T;
    dy[7] = P.ta * T;
}

__global__ __launch_bounds__(32) void sidarthe_rk4_kernel(
    const float* alpha, const float* beta, const float* gamma_, const float* delta,
    const float* epsilon, const float* theta, const float* zeta, const float* eta,
    const float* mu, const float* nu, const float* tau, const float* lambda_,
    const float* rho, const float* kappa, const float* xi, const float* sigma,
    const float* y0, const float* I0, const float* E0, const float* H0,
    float* __restrict__ out)
{
    __shared__ __align__(16) float ring[2][SLOT_ELEMS];

    if (threadIdx.x != 0) return;   // one serial lane; TDM/scalar waits ignore EXEC

    Params P;
    P.a = alpha[0];   P.b = beta[0];    P.g = gamma_[0];  P.d = delta[0];
    P.eps = epsilon[0]; P.th = theta[0]; P.ze = zeta[0];  P.et = eta[0];
    P.m = mu[0];      P.n = nu[0];      P.ta = tau[0];    P.la = lambda_[0];
    P.rh = rho[0];    P.ka = kappa[0];  P.x = xi[0];      P.si = sigma[0];
    P.cD = P.et + P.rh;
    P.cA = P.th + P.m + P.ka;
    P.cR = P.n + P.x;
    P.cT = P.si + P.ta;
    P.cI = P.eps + P.ze + P.la;

    float y[NSTATE];
    y[0] = y0[0]; y[1] = y0[1]; y[2] = y0[2]; y[3] = y0[3]; y[4] = y0[4];
    y[5] = I0[0]; y[6] = E0[0]; y[7] = H0[0];

    int slot = 0, row = 0, base = 0;

    // TDM flush: store nrows*8 contiguous f32 from LDS slot to out[rowBase*8..]
    auto flush = [&](int s, int rowBase, int nrows) {
        asm volatile("s_wait_dscnt 0" ::: "memory");  // LDS writes visible to TDM
        const unsigned nElem  = (unsigned)(nrows * NSTATE);          // <= 4096
        const unsigned ldsOff = (unsigned)(unsigned long long)(&ring[s][0]);
        const unsigned long long gaddr =
            (unsigned long long)(out + (unsigned long long)rowBase * NSTATE);

        u32x4 g0;
        g0[0] = 1u;                                        // count=1, user desc
        g0[1] = ldsOff;                                    // lds_addr (bytes)
        g0[2] = (unsigned)(gaddr & 0xFFFFFFFFull);         // global_addr[31:0]
        g0[3] = (unsigned)((gaddr >> 32) & 0x1FFFFFFull)   // global_addr[56:32]
              | (2u << 30);                                // type = 2 ("image")

        i32x8 g1;
        g1[0] = (int)(2u << 16);                           // data_size = 4B
        g1[1] = (int)((nElem & 0xFFFFu) << 16);            // tensor_dim0[15:0]
        g1[2] = (int)((nElem >> 16) | (1u << 16));         // tensor_dim0[31:16] | tensor_dim1=1
        g1[3] = (int)(nElem << 16);                        // tile_dim0 = nElem (1-D tile)
        g1[4] = 0;                                         // tile_dim1=0, tile_dim2=0
        g1[5] = (int)nElem;                                // tensor_dim0_stride
        g1[6] = 0;
        g1[7] = 0;

        i32x4 z4 = {0, 0, 0, 0};                           // groups 2/3 unused (2-D max)
        i32x8 z8 = {0, 0, 0, 0, 0, 0, 0, 0};
        __builtin_amdgcn_tensor_store_from_lds(g0, g1, z4, z4, z8, 0);
    };

    auto push = [&](const float* v) {
        if (row == 0)                                      // reusing a slot: its prior
            __builtin_amdgcn_s_wait_tensorcnt(1);          // TDM store must be done
        float4* dst = (float4*)&ring[slot][row * NSTATE];
        dst[0] = make_float4(v[0], v[1], v[2], v[3]);
        dst[1] = make_float4(v[4], v[5], v[6], v[7]);
        if (++row == SLOT_ROWS) {
            flush(slot, base, SLOT_ROWS);
            base += SLOT_ROWS;
            slot ^= 1;
            row = 0;
        }
    };

    push(y);                                               // row 0 = y_init

    const float third = 1.0f / 3.0f;
    for (int t = 0; t < STEPS; ++t) {
        float k1[NSTATE], k2[NSTATE], k3[NSTATE], k4[NSTATE], tmp[NSTATE];

        rhs(y, P, k1);
        #pragma unroll
        for (int i = 0; i < NSTATE; ++i) tmp[i] = y[i] + third * k1[i];
        rhs(tmp, P, k2);
        #pragma unroll
        for (int i = 0; i < NSTATE; ++i) tmp[i] = y[i] + (k2[i] - third * k1[i]);
        rhs(tmp, P, k3);
        #pragma unroll
        for (int i = 0; i < NSTATE; ++i) tmp[i] = y[i] + (k1[i] - k2[i] + k3[i]);
        rhs(tmp, P, k4);
        #pragma unroll
        for (int i = 0; i < NSTATE; ++i)
            y[i] = y[i] + 0.125f * (k1[i] + 3.0f * (k2[i] + k3[i]) + k4[i]);

        push(y);
    }

    if (row > 0) flush(slot, base, row);                   // tail (20001 = 39*512+33)
    __builtin_amdgcn_s_wait_tensorcnt(0);
}

extern "C" void kernel_launch(void* const* d_in, const int* in_sizes, int n_in,
                              void* d_out, int out_size, void* d_ws, size_t ws_size,
                              hipStream_t stream) {
    (void)in_sizes; (void)n_in; (void)out_size; (void)d_ws; (void)ws_size;
    sidarthe_rk4_kernel<<<1, 32, 0, stream>>>(
        (const float*)d_in[0],  (const float*)d_in[1],
        (const float*)d_in[2],  (const float*)d_in[3],
        (const float*)d_in[4],  (const float*)d_in[5],
        (const float*)d_in[6],  (const float*)d_in[7],
        (const float*)d_in[8],  (const float*)d_in[9],
        (const float*)d_in[10], (const float*)d_in[11],
        (const float*)d_in[12], (const float*)d_in[13],
        (const float*)d_in[14], (const float*)d_in[15],
        (const float*)d_in[16],                       // y0 (5)
        (const float*)d_in[17],                       // I0
        (const float*)d_in[18],                       // E0
        (const float*)d_in[19],                       // H0
        (float*)d_out);
}